// SimpleModel_17712445129203
// MI455X (gfx1250) — compile-verified
//
#include <hip/hip_runtime.h>

typedef float v2f __attribute__((ext_vector_type(2)));
typedef float v8f __attribute__((ext_vector_type(8)));

// -----------------------------------------------------------------------------
// Y[M,HD] = X[M,K] @ W[K,HD] + b[HD]
// M multiple of 16, K multiple of 4, HD == 128 (8 N-tiles = 8 waves per block).
// One wave computes one 16x16 f32 tile with V_WMMA_F32_16X16X4_F32.
// A 16x4 layout: lanes 0-15 hold M=lane, VGPR0=K+0, VGPR1=K+1;
//                lanes 16-31 hold M=lane-16, VGPR0=K+2, VGPR1=K+3.
// B 4x16 layout mirrors A with N instead of M (row striped across lanes).
// C/D: VGPR r -> M = r + 8*(lane>=16), N = lane&15.
// -----------------------------------------------------------------------------
__global__ __launch_bounds__(256) void gemm_bias_f32(
    const float* __restrict__ X, const float* __restrict__ W,
    const float* __restrict__ b, float* __restrict__ Y,
    int K, int HD)
{
    const int lane  = threadIdx.x & 31;
    const int ntile = threadIdx.x >> 5;      // wave id -> N tile (HD/16 == 8)
    const int mtile = blockIdx.x;
    const int half  = lane >> 4;             // 0: lanes 0-15, 1: lanes 16-31
    const int l16   = lane & 15;

    const int row = mtile * 16 + l16;        // A row for this lane
    const int col = ntile * 16 + l16;        // B/C/D column for this lane

    v8f acc = {};
    for (int k = 0; k < K; k += 4) {
        // A fragment: contiguous pair {K = k+2*half, k+2*half+1} of row `row`
        v2f a = *(const v2f*)(X + (size_t)row * K + k + 2 * half);
        // B fragment: rows k+2*half and k+2*half+1, column `col`
        v2f bb;
        bb.x = W[(size_t)(k + 2 * half)     * HD + col];
        bb.y = W[(size_t)(k + 2 * half + 1) * HD + col];
        acc = __builtin_amdgcn_wmma_f32_16x16x4_f32(
            /*neg_a=*/false, a, /*neg_b=*/false, bb,
            /*c_mod=*/(short)0, acc, /*reuse_a=*/false, /*reuse_b=*/false);
    }

    const float bias = b[col];
#pragma unroll
    for (int r = 0; r < 8; ++r) {
        const int orow = mtile * 16 + r + 8 * half;
        Y[(size_t)orow * HD + col] = acc[r] + bias;
    }
}

// -----------------------------------------------------------------------------
// hacc[dst[e], :] += hin[src[e], :] * (deg ? 1/deg[src[e]] : 1)
// One wave per edge; each lane moves a float4 (32 lanes * 4 = 128 = HD).
// Accumulator region (51.2 MB) is L2-resident -> atomics stay in L2.
// -----------------------------------------------------------------------------
__global__ __launch_bounds__(256) void edge_scatter_add(
    const float* __restrict__ hin, float* __restrict__ hacc,
    const int* __restrict__ src, const int* __restrict__ dst,
    const float* __restrict__ deg, int E, int HD)
{
    const int lane = threadIdx.x & 31;
    const long long e = ((long long)blockIdx.x * blockDim.x + threadIdx.x) >> 5;
    if (e >= E) return;

    const int s = src[e];
    const int d = dst[e];
    const float scale = deg ? (1.0f / deg[s]) : 1.0f;

    const float4 v = *(const float4*)(hin + (size_t)s * HD + lane * 4);
    float* o = hacc + (size_t)d * HD + lane * 4;
    atomicAdd(o + 0, v.x * scale);
    atomicAdd(o + 1, v.y * scale);
    atomicAdd(o + 2, v.z * scale);
    atomicAdd(o + 3, v.w * scale);
}

// -----------------------------------------------------------------------------
// feat[j] += sum over a block's row chunk of h[r, j]   (blockDim == HD == 128)
// -----------------------------------------------------------------------------
__global__ __launch_bounds__(128) void col_sum(
    const float* __restrict__ h, float* __restrict__ feat,
    int N, int HD, int rowsPerBlock)
{
    const int j = threadIdx.x;
    long long r0 = (long long)blockIdx.x * rowsPerBlock;
    long long r1 = r0 + rowsPerBlock;
    if (r1 > N) r1 = N;
    float s = 0.0f;
    for (long long r = r0; r < r1; ++r) s += h[r * HD + j];
    atomicAdd(feat + j, s);
}

// -----------------------------------------------------------------------------
// out = softmax((feat/N) @ Wp + bp), C == 2. One wave.
// -----------------------------------------------------------------------------
__global__ __launch_bounds__(32) void predict_softmax(
    const float* __restrict__ feat, const float* __restrict__ Wp,
    const float* __restrict__ bp, float* __restrict__ out,
    int HD, int C, float invN)
{
    const int lane = threadIdx.x;
    float l0 = 0.0f, l1 = 0.0f;
    for (int j = lane; j < HD; j += 32) {
        const float f = feat[j] * invN;
        l0 += f * Wp[j * C + 0];
        l1 += f * Wp[j * C + 1];
    }
#pragma unroll
    for (int off = 16; off > 0; off >>= 1) {
        l0 += __shfl_down(l0, off, 32);
        l1 += __shfl_down(l1, off, 32);
    }
    if (lane == 0) {
        l0 += bp[0];
        l1 += bp[1];
        const float m  = fmaxf(l0, l1);
        const float e0 = __expf(l0 - m);
        const float e1 = __expf(l1 - m);
        const float inv = 1.0f / (e0 + e1);
        out[0] = e0 * inv;
        out[1] = e1 * inv;
    }
}

// -----------------------------------------------------------------------------
// Inputs (setup_inputs order):
//  0 node_feat [N,NF]  1 edge_feat (dead)  2 base_data (dead)
//  3 src [E]           4 dst [E]           5 degree [N,1]
//  6 Wn [NF,HD]  7 bn [HD]  8 We (dead)  9 be (dead) 10 Wg (dead) 11 bg (dead)
// 12 Wgcn [HD,HD] 13 bgcn [HD] 14 Wp [HD,C] 15 bp [C]
// -----------------------------------------------------------------------------
extern "C" void kernel_launch(void* const* d_in, const int* in_sizes, int n_in,
                              void* d_out, int out_size, void* d_ws, size_t ws_size,
                              hipStream_t stream)
{
    const float* node_feat = (const float*)d_in[0];
    const int*   src       = (const int*)  d_in[3];
    const int*   dst       = (const int*)  d_in[4];
    const float* degree    = (const float*)d_in[5];
    const float* Wn        = (const float*)d_in[6];
    const float* bn        = (const float*)d_in[7];
    const float* Wgcn      = (const float*)d_in[12];
    const float* bgcn      = (const float*)d_in[13];
    const float* Wp        = (const float*)d_in[14];
    const float* bp        = (const float*)d_in[15];
    float*       out       = (float*)d_out;

    const int HD = in_sizes[7];          // 128
    const int NF = in_sizes[6] / HD;     // 256
    const int Nn = in_sizes[0] / NF;     // 100000 (multiple of 16)
    const int E  = in_sizes[3];          // 1600000
    const int C  = in_sizes[15];         // 2

    const size_t hbytes = (size_t)Nn * HD * sizeof(float);
    float* hA   = (float*)d_ws;                             // ping
    float* hB   = (float*)((char*)d_ws + hbytes);           // pong
    float* feat = (float*)((char*)d_ws + 2 * hbytes);       // [HD]

    const int mtiles = Nn / 16;                  // 6250
    const int egrid  = (E + 7) / 8;              // 8 waves (edges) per 256-thr block

    // 1) hA = node_feat @ Wn + bn
    gemm_bias_f32<<<mtiles, 256, 0, stream>>>(node_feat, Wn, bn, hA, NF, HD);

    // 2) Node_feat_fusion: hB = segment_sum(hA[src], dst)
    hipMemsetAsync(hB, 0, hbytes, stream);
    edge_scatter_add<<<egrid, 256, 0, stream>>>(hA, hB, src, dst, nullptr, E, HD);

    // 3) 3 GCN layers with shared weight:
    //    agg = segment_sum(h[src]/deg[src], dst) + h ; h = agg @ Wgcn + bgcn
    for (int l = 0; l < 3; ++l) {
        hipMemcpyAsync(hA, hB, hbytes, hipMemcpyDeviceToDevice, stream); // agg := h
        edge_scatter_add<<<egrid, 256, 0, stream>>>(hB, hA, src, dst, degree, E, HD);
        gemm_bias_f32<<<mtiles, 256, 0, stream>>>(hA, Wgcn, bgcn, hB, HD, HD);
    }

    // 4) feat = column-sum(h); 5) softmax((feat/N) @ Wp + bp)
    hipMemsetAsync(feat, 0, HD * sizeof(float), stream);
    const int rowsPerBlock = 256;
    const int nblk = (Nn + rowsPerBlock - 1) / rowsPerBlock;
    col_sum<<<nblk, HD, 0, stream>>>(hB, feat, Nn, HD, rowsPerBlock);
    predict_softmax<<<1, 32, 0, stream>>>(feat, Wp, bp, out, HD, C, 1.0f / (float)Nn);
}